// SetAbstraction_77025943486586
// MI455X (gfx1250) — compile-verified
//
#include <hip/hip_runtime.h>
#include <hip/hip_bf16.h>
#include <math.h>

typedef __attribute__((ext_vector_type(2))) float v2f;
typedef __attribute__((ext_vector_type(8))) float v8f;

#define CB 8
#define CN 16384
#define CF 32
#define CS 2048
#define CK 32
#define COUT 128

__device__ __forceinline__ v8f wmma_f32(v2f a, v2f b, v8f c) {
  // D = A(16x4,f32) * B(4x16,f32) + C(16x16,f32)
  return __builtin_amdgcn_wmma_f32_16x16x4_f32(false, a, false, b, (short)0, c,
                                               false, false);
}

// Async 16B copy global -> LDS (GV mode: 64-bit address in VGPR pair).
// Low 32 bits of a flat shared-memory pointer are the LDS byte offset.
__device__ __forceinline__ void async_copy_b128(const float* lds_dst,
                                                const float* gsrc) {
  unsigned ldsoff = (unsigned)(uintptr_t)lds_dst;
  unsigned long long ga = (unsigned long long)(uintptr_t)gsrc;
  asm volatile("global_load_async_to_lds_b128 %0, %1, off"
               :: "v"(ldsoff), "v"(ga) : "memory");
}

__device__ __forceinline__ void wait_async0() {
  asm volatile("s_wait_asynccnt 0" ::: "memory");
}

// ---------------------------------------------------------------- x_sq kernel
__global__ void xsq_kernel(const float* __restrict__ x, float* __restrict__ xsq) {
  int i = blockIdx.x * blockDim.x + threadIdx.x;   // b*N + n
  if (i >= CB * CN) return;
  const float* p = x + (size_t)i * CF;
  float s = 0.f;
#pragma unroll
  for (int f = 0; f < CF; ++f) s += p[f] * p[f];
  xsq[i] = s;
}

// -------------------------------------- sampled gather -> out [B, F, S] part
__global__ void sampled_kernel(const float* __restrict__ x,
                               const int* __restrict__ sidx,
                               float* __restrict__ out2) {
  int i = blockIdx.x * blockDim.x + threadIdx.x;   // b*S + s
  if (i >= CB * CS) return;
  int b = i / CS, s = i - b * CS;
  int n = sidx[i];
  const float* p = x + ((size_t)b * CN + n) * CF;
  float* o = out2 + (size_t)b * CF * CS + s;
#pragma unroll
  for (int f = 0; f < CF; ++f) o[(size_t)f * CS] = p[f];
}

// ------------------------------------------------------------------ KNN kernel
// 1 workgroup (128 thr = 4 waves) handles 16 sampled rows; streams N in 64-col
// chunks double-buffered via GLOBAL_LOAD_ASYNC_TO_LDS_B128 (ASYNCcnt); cross
// products via V_WMMA_F32_16X16X4_F32; per-row top-32 with lane k == entry k.
#define TS 16
#define TC 64
#define XS 36   // padded LDS row stride (16B-aligned rows, conflict-free reads)
__global__ __launch_bounds__(128) void knn_kernel(const float* __restrict__ x,
                                                  const int* __restrict__ sidx,
                                                  const float* __restrict__ xsq,
                                                  int* __restrict__ nbr) {
  __shared__ float ssh[TS * 33];        // sampled rows [row][f]
  __shared__ float ssq[TS];
  __shared__ int   sIdxSh[TS];
  __shared__ float xsh[2][TC * XS];     // x chunk [col][f], double buffered
  __shared__ float xq[2][TC];
  __shared__ float dsh[TS * (TC + 1)];  // distance tile [row][col]

  const int b  = blockIdx.x / (CS / TS);
  const int sb = (blockIdx.x % (CS / TS)) * TS;
  const int t  = threadIdx.x;
  const int w  = t >> 5;
  const int lane = t & 31;
  const int m  = lane & 15;
  const int hi = lane >> 4;
  const int koff = hi * 2;

  auto issue_chunk = [&](int n0, int buf) {
    // 64 cols x 32 floats = 512 x b128; 128 threads x 4
#pragma unroll
    for (int j = 0; j < 4; ++j) {
      int i = t + 128 * j;
      int c = i >> 3;            // 0..63
      int f = (i & 7) * 4;       // 0,4,...,28
      async_copy_b128(&xsh[buf][c * XS + f],
                      &x[((size_t)b * CN + n0 + c) * CF + f]);
    }
    if (t < TC) xq[buf][t] = xsq[b * CN + n0 + t];
  };

  issue_chunk(0, 0);             // prefetch first chunk

  if (t < TS) sIdxSh[t] = sidx[b * CS + sb + t];
  __syncthreads();
#pragma unroll
  for (int j = 0; j < 4; ++j) {
    int i = t + 128 * j;
    int r = i >> 5, f = i & 31;
    ssh[r * 33 + f] = x[((size_t)b * CN + sIdxSh[r]) * CF + f];
  }
  __syncthreads();
  if (t < TS) {
    float s = 0.f;
#pragma unroll
    for (int f = 0; f < CF; ++f) { float v = ssh[t * 33 + f]; s += v * v; }
    ssq[t] = s;
  }

  // hoist loop-invariant A operand (sampled tile) into registers
  v2f areg[8];
#pragma unroll
  for (int k0 = 0; k0 < CF; k0 += 4) {
    areg[k0 >> 2].x = ssh[m * 33 + k0 + koff];
    areg[k0 >> 2].y = ssh[m * 33 + k0 + koff + 1];
  }

  // top-K state: lane k = entry k, 4 rows per wave
  float hd[4];  int hix[4];  float cmx[4];
#pragma unroll
  for (int rl = 0; rl < 4; ++rl) { hd[rl] = 3.4e38f; hix[rl] = 0; cmx[rl] = 3.4e38f; }

  for (int n0 = 0; n0 < CN; n0 += TC) {
    const int buf = (n0 / TC) & 1;
    wait_async0();               // my async copies for this chunk are done
    __syncthreads();             // everyone's are done; prev top-K reads done

    // wave w computes cross for cols [16w, 16w+16)
    v8f acc = {};
#pragma unroll
    for (int k0 = 0; k0 < CF; k0 += 4) {
      v2f bb;
      int col = 16 * w + m;
      bb.x = xsh[buf][col * XS + k0 + koff];
      bb.y = xsh[buf][col * XS + k0 + koff + 1];
      acc = wmma_f32(areg[k0 >> 2], bb, acc);
    }
#pragma unroll
    for (int v = 0; v < 8; ++v) {
      int row = v + 8 * hi;
      float d = ssq[row] + xq[buf][16 * w + m] - 2.f * acc[v];
      dsh[row * (TC + 1) + 16 * w + m] = d;
    }

    if (n0 + TC < CN) issue_chunk(n0 + TC, buf ^ 1);   // overlap with top-K
    __syncthreads();

    // wave w maintains top-32 for rows [4w, 4w+4)
#pragma unroll
    for (int rl = 0; rl < 4; ++rl) {
      int row = 4 * w + rl;
      float cm = cmx[rl];
      for (int j = 0; j < TC; ++j) {
        float cand = dsh[row * (TC + 1) + j];      // uniform across wave
        if (cand < cm) {
          float red = hd[rl];
          red = fmaxf(red, __shfl_xor(red, 16, 32));
          red = fmaxf(red, __shfl_xor(red, 8, 32));
          red = fmaxf(red, __shfl_xor(red, 4, 32));
          red = fmaxf(red, __shfl_xor(red, 2, 32));
          red = fmaxf(red, __shfl_xor(red, 1, 32));
          unsigned long long msk = __ballot(hd[rl] == red);
          int ml = __ffsll(msk) - 1;
          if (lane == ml) { hd[rl] = cand; hix[rl] = n0 + j; }
          float nm = hd[rl];
          nm = fmaxf(nm, __shfl_xor(nm, 16, 32));
          nm = fmaxf(nm, __shfl_xor(nm, 8, 32));
          nm = fmaxf(nm, __shfl_xor(nm, 4, 32));
          nm = fmaxf(nm, __shfl_xor(nm, 2, 32));
          nm = fmaxf(nm, __shfl_xor(nm, 1, 32));
          cm = nm;
        }
      }
      cmx[rl] = cm;
    }
  }
  __syncthreads();

#pragma unroll
  for (int rl = 0; rl < 4; ++rl) {
    int row = sb + 4 * w + rl;
    nbr[((size_t)b * CS + row) * CK + lane] = hix[rl];
  }
}

// ------------------------------------------------------- MLP + max-pool kernel
// 1 workgroup (128 thr = 4 waves) per (b,s). Async-gather neighbors to LDS,
// two WMMA GEMMs, neighbor max via shfl.
__global__ __launch_bounds__(128) void mlp_kernel(const float* __restrict__ x,
                                                  const int* __restrict__ nbr,
                                                  const float* __restrict__ W1,
                                                  const float* __restrict__ b1,
                                                  const float* __restrict__ W2,
                                                  const float* __restrict__ b2,
                                                  float* __restrict__ out) {
  __shared__ float gsh[32 * XS];    // gathered neighbor features [k][f]
  __shared__ float hsh[32 * 132];   // hidden activations [k][o]
  __shared__ int   nIdx[32];

  const int bs = blockIdx.x;
  const int b = bs / CS, s = bs - b * CS;
  const int t = threadIdx.x;
  const int w = t >> 5;
  const int lane = t & 31;
  const int m = lane & 15;
  const int hi = lane >> 4;
  const int koff = hi * 2;

  if (t < 32) nIdx[t] = nbr[(size_t)bs * CK + t];
  __syncthreads();
  // 32 rows x 32 floats = 256 x b128; 128 threads x 2, async to LDS
#pragma unroll
  for (int j = 0; j < 2; ++j) {
    int i = t + 128 * j;
    int r = i >> 3;              // 0..31
    int f = (i & 7) * 4;
    async_copy_b128(&gsh[r * XS + f],
                    &x[((size_t)b * CN + nIdx[r]) * CF + f]);
  }
  wait_async0();
  __syncthreads();

  // MLP1: h = relu(G[32x32] * W1^T[32x128] + b1); wave w owns 2 col tiles
  for (int tt = 0; tt < 2; ++tt) {
    int ob = (2 * w + tt) * 16;
    float bias = b1[ob + m];
    for (int mt = 0; mt < 2; ++mt) {
      v8f acc = {};
#pragma unroll
      for (int k0 = 0; k0 < CF; k0 += 4) {
        v2f a, bb;
        a.x  = gsh[(mt * 16 + m) * XS + k0 + koff];
        a.y  = gsh[(mt * 16 + m) * XS + k0 + koff + 1];
        bb.x = W1[(size_t)(ob + m) * CF + k0 + koff];
        bb.y = W1[(size_t)(ob + m) * CF + k0 + koff + 1];
        acc = wmma_f32(a, bb, acc);
      }
#pragma unroll
      for (int v = 0; v < 8; ++v) {
        int row = mt * 16 + v + 8 * hi;
        hsh[row * 132 + ob + m] = fmaxf(acc[v] + bias, 0.f);
      }
    }
  }
  __syncthreads();

  // MLP2: h2 = h[32x128] * W2^T[128x128] + b2; then max over 32 neighbors
  for (int tt = 0; tt < 2; ++tt) {
    int pb = (2 * w + tt) * 16;
    v8f acc0 = {}, acc1 = {};
    for (int k0 = 0; k0 < COUT; k0 += 4) {
      v2f bb, a0, a1;
      bb.x = W2[(size_t)(pb + m) * COUT + k0 + koff];
      bb.y = W2[(size_t)(pb + m) * COUT + k0 + koff + 1];
      a0.x = hsh[m * 132 + k0 + koff];
      a0.y = hsh[m * 132 + k0 + koff + 1];
      a1.x = hsh[(16 + m) * 132 + k0 + koff];
      a1.y = hsh[(16 + m) * 132 + k0 + koff + 1];
      acc0 = wmma_f32(a0, bb, acc0);
      acc1 = wmma_f32(a1, bb, acc1);
    }
    float mx = -3.4e38f;
#pragma unroll
    for (int v = 0; v < 8; ++v) mx = fmaxf(mx, fmaxf(acc0[v], acc1[v]));
    mx = fmaxf(mx, __shfl_xor(mx, 16, 32));   // merge the two row-halves
    mx += b2[pb + m];
    if (hi == 0)
      out[(size_t)b * COUT * CS + (size_t)(pb + m) * CS + s] = mx;
  }
}

// ------------------------------------------------------------------- launcher
extern "C" void kernel_launch(void* const* d_in, const int* in_sizes, int n_in,
                              void* d_out, int out_size, void* d_ws, size_t ws_size,
                              hipStream_t stream) {
  const float* x   = (const float*)d_in[0];
  const int* sidx  = (const int*)d_in[1];
  const float* W1  = (const float*)d_in[2];
  const float* b1  = (const float*)d_in[3];
  const float* W2  = (const float*)d_in[4];
  const float* b2  = (const float*)d_in[5];
  float* out = (float*)d_out;                       // [B,OUT,S] then [B,F,S]
  float* out_sampled = out + (size_t)CB * COUT * CS;

  float* xsq = (float*)d_ws;                                  // B*N floats
  int* nbr = (int*)((char*)d_ws + sizeof(float) * CB * CN);   // B*S*K ints

  xsq_kernel<<<(CB * CN + 255) / 256, 256, 0, stream>>>(x, xsq);
  sampled_kernel<<<(CB * CS + 255) / 256, 256, 0, stream>>>(x, sidx, out_sampled);
  knn_kernel<<<CB * (CS / TS), 128, 0, stream>>>(x, sidx, xsq, nbr);
  mlp_kernel<<<CB * CS, 128, 0, stream>>>(x, nbr, W1, b1, W2, b2, out);
}